// FU_36404142801582
// MI455X (gfx1250) — compile-verified
//
#include <hip/hip_runtime.h>
#include <hip/hip_bf16.h>

// Problem constants (from reference)
#define Bn   8
#define Cn   64
#define Hn   128
#define Wn   128
#define Gn   4
#define Cg   16      // channels per group (= contraction sub-dim)
#define Cog  16      // output channels per group (= WMMA M)
#define Knum 9       // 3x3 taps
#define HWn  (Hn * Wn)
#define OCoff 18     // offset-conv output channels (2*K)

typedef __attribute__((ext_vector_type(2))) float v2f;
typedef __attribute__((ext_vector_type(8))) float v8f;

// ---------------------------------------------------------------------------
// Kernel 1: offsets = conv3x3(x, offset_w) + offset_b
// grid = (HW/256, B*18), block = 256. Weight indices are block-uniform.
// ---------------------------------------------------------------------------
__global__ void __launch_bounds__(256) FU_offset_conv_kernel(
    const float* __restrict__ x,      // (B,64,H,W)
    const float* __restrict__ ow,     // (18,64,3,3)
    const float* __restrict__ obias,  // (18,)
    float* __restrict__ offs)         // (B,18,H,W)
{
    const int boc = blockIdx.y;           // b*18 + oc
    const int oc  = boc % OCoff;
    const int pix = blockIdx.x * 256 + threadIdx.x;
    const int ho  = pix >> 7;             // / W (W = 128)
    const int wo  = pix & (Wn - 1);

    const float* xb = x + (size_t)(boc / OCoff) * Cn * HWn;
    const float* wb = ow + (size_t)oc * Cn * 9;

    float acc = obias[oc];
    for (int c = 0; c < Cn; ++c) {
        const float* xc = xb + (size_t)c * HWn;
        const float* wc = wb + c * 9;
        #pragma unroll
        for (int t = 0; t < 9; ++t) {
            const int ky = t / 3;
            const int kx = t - ky * 3;
            const int y  = ho + ky - 1;
            const int xx = wo + kx - 1;
            const float v = ((unsigned)y < (unsigned)Hn &&
                             (unsigned)xx < (unsigned)Wn)
                            ? xc[y * Wn + xx] : 0.0f;
            acc = fmaf(v, wc[t], acc);
        }
    }
    offs[(size_t)boc * HWn + pix] = acc;
}

// ---------------------------------------------------------------------------
// Kernel 2: fused bilinear deform-sample + grouped 16x16 contraction via
// V_WMMA_F32_16X16X4_F32.  One wave = one (16 outch x 16 pixel) tile; one
// 256-thread block = 8 waves = one full image row (W=128) for one (b,g).
// grid = (H, B*G), block = 256.
// ---------------------------------------------------------------------------
__global__ void __launch_bounds__(256) FU_deform_wmma_kernel(
    const float* __restrict__ skip,   // (B,64,H,W)
    const float* __restrict__ offs,   // (B,18,H,W)
    const float* __restrict__ dw,     // (64,16,3,3)
    float* __restrict__ out)          // (B,64,H,W)
{
    // LDS weight tile for this group, laid out A-fragment friendly:
    // w_lds[o*146 + k*16 + c] = dw[g*16+o][c][ky][kx];  stride 146 keeps
    // 8B alignment for ds_load_b64 and breaks 64-bank alignment.
    __shared__ float w_lds[Cog * 146];

    const int bg = blockIdx.y;
    const int b  = bg / Gn;
    const int g  = bg - b * Gn;
    const int ho = blockIdx.x;

    for (int l = threadIdx.x; l < Cog * (Cg * Knum); l += 256) {
        const int o   = l / 144;
        const int rem = l - o * 144;
        const int k   = rem >> 4;
        const int c   = rem & 15;
        w_lds[o * 146 + rem] = dw[((size_t)(g * Cog + o) * Cg + c) * 9 + k];
    }
    __syncthreads();

    const int lane = threadIdx.x & 31;
    const int wave = threadIdx.x >> 5;   // 0..7
    const int hi   = lane >> 4;          // lane-half selects K-subrows 0/1 vs 2/3
    const int nidx = lane & 15;          // WMMA: A-row M, B-col N, pixel-in-tile
    const int px_i = wave * 16 + nidx;   // pixel x coordinate (0..127)

    const float* sgb = skip + ((size_t)b * Cn + g * Cg) * HWn;
    const float* ob  = offs + (size_t)b * OCoff * HWn + ho * Wn + px_i;

    v8f acc = {};

    #pragma unroll
    for (int k = 0; k < Knum; ++k) {
        const int ky = k / 3;
        const int kx = k - ky * 3;

        // learned offsets for this tap at this pixel (channels 2k, 2k+1)
        const float offy = ob[(size_t)(2 * k)     * HWn];
        const float offx = ob[(size_t)(2 * k + 1) * HWn];

        const float py  = (float)(ho   - 1 + ky) + offy;
        const float pxx = (float)(px_i - 1 + kx) + offx;
        const float y0f = floorf(py);
        const float x0f = floorf(pxx);
        const float wy  = py  - y0f;
        const float wx  = pxx - x0f;
        const int y0 = (int)y0f, x0 = (int)x0f;
        const int y1 = y0 + 1,   x1 = x0 + 1;

        const bool vy0 = (unsigned)y0 < (unsigned)Hn;
        const bool vy1 = (unsigned)y1 < (unsigned)Hn;
        const bool vx0 = (unsigned)x0 < (unsigned)Wn;
        const bool vx1 = (unsigned)x1 < (unsigned)Wn;

        // validity folded into the bilinear weights (matches reference mask)
        const float w00 = (vy0 && vx0) ? (1.0f - wy) * (1.0f - wx) : 0.0f;
        const float w01 = (vy0 && vx1) ? (1.0f - wy) * wx          : 0.0f;
        const float w10 = (vy1 && vx0) ? wy * (1.0f - wx)          : 0.0f;
        const float w11 = (vy1 && vx1) ? wy * wx                   : 0.0f;

        const int cy0 = min(max(y0, 0), Hn - 1);
        const int cy1 = min(max(y1, 0), Hn - 1);
        const int cx0 = min(max(x0, 0), Wn - 1);
        const int cx1 = min(max(x1, 0), Wn - 1);
        const int i00 = cy0 * Wn + cx0;
        const int i01 = cy0 * Wn + cx1;
        const int i10 = cy1 * Wn + cx0;
        const int i11 = cy1 * Wn + cx1;

        // contraction over Cg=16 channels in K=4 WMMA steps
        #pragma unroll
        for (int c0 = 0; c0 < Cg; c0 += 4) {
            const int c = c0 + hi * 2;              // this lane feeds rows c, c+1
            const float* s0 = sgb + (size_t)c * HWn;
            const float* s1 = s0 + HWn;

            v2f bfrag;
            bfrag.x = w00 * s0[i00] + w01 * s0[i01] + w10 * s0[i10] + w11 * s0[i11];
            bfrag.y = w00 * s1[i00] + w01 * s1[i01] + w10 * s1[i10] + w11 * s1[i11];

            const v2f afrag =
                *(const v2f*)&w_lds[nidx * 146 + k * 16 + c];

            acc = __builtin_amdgcn_wmma_f32_16x16x4_f32(
                      /*neg_a=*/false, afrag,
                      /*neg_b=*/false, bfrag,
                      /*c_mod=*/(short)0, acc,
                      /*reuse_a=*/false, /*reuse_b=*/false);
        }
    }

    // C/D 16x16 f32 layout: VGPR r -> M = r + 8*hi, N = lane&15
    float* og = out + ((size_t)b * Cn + g * Cg) * HWn + ho * Wn + px_i;
    #pragma unroll
    for (int r = 0; r < 8; ++r) {
        og[(size_t)(r + hi * 8) * HWn] = acc[r];
    }
}

// ---------------------------------------------------------------------------
extern "C" void kernel_launch(void* const* d_in, const int* in_sizes, int n_in,
                              void* d_out, int out_size, void* d_ws, size_t ws_size,
                              hipStream_t stream) {
    const float* x     = (const float*)d_in[0];
    const float* skip  = (const float*)d_in[1];
    const float* offw  = (const float*)d_in[2];
    const float* offb  = (const float*)d_in[3];
    const float* dwt   = (const float*)d_in[4];
    float*       out   = (float*)d_out;
    float*       offs  = (float*)d_ws;   // (B,18,H,W) fp32 = 9.4 MB scratch

    // Stage 1: offset conv
    dim3 g1(HWn / 256, Bn * OCoff);
    FU_offset_conv_kernel<<<g1, 256, 0, stream>>>(x, offw, offb, offs);

    // Stage 2: fused deformable sample + grouped WMMA contraction
    dim3 g2(Hn, Bn * Gn);
    FU_deform_wmma_kernel<<<g2, 256, 0, stream>>>(skip, offs, dwt, out);
}